// HCLoss_62835371540980
// MI455X (gfx1250) — compile-verified
//
#include <hip/hip_runtime.h>
#include <hip/hip_bf16.h>
#include <stdint.h>

typedef float v2f __attribute__((ext_vector_type(2)));
typedef float v8f __attribute__((ext_vector_type(8)));
typedef int   v4i_vs __attribute__((vector_size(4 * sizeof(int))));  // matches builtin proto

#define B_ROWS     4096
#define DIMS       128
#define EACH_CL    4
#define TILE       16
#define NWAVES     8
#define NTILES     (B_ROWS / TILE)      // 256
#define NSPLIT     4                    // column-space split for occupancy
#define TT_ITERS   (NTILES / NSPLIT)    // 64 col tiles per block
#define BSTRIDE    132                  // floats; 528B rows: 16B aligned, bank-friendly

// ---------------------------------------------------------------------------
// Async global->LDS helpers (ASYNCcnt path), with inline-asm fallbacks.
// ---------------------------------------------------------------------------
__device__ __forceinline__ void async_ld_b128(const float* g, float* l) {
#if __has_builtin(__builtin_amdgcn_global_load_async_to_lds_b128)
    __builtin_amdgcn_global_load_async_to_lds_b128(
        (__attribute__((address_space(1))) v4i_vs*)(uintptr_t)g,
        (__attribute__((address_space(3))) v4i_vs*)(uintptr_t)l,
        0, 0);
#else
    // VDST = per-lane LDS byte offset, VADDR = 64-bit global address.
    unsigned ldsOff = (unsigned)(uintptr_t)l;   // low 32 bits = LDS offset
    asm volatile("global_load_async_to_lds_b128 %0, %1, off"
                 :: "v"(ldsOff), "v"(g) : "memory");
#endif
}

__device__ __forceinline__ void wait_async_le2() {
#if __has_builtin(__builtin_amdgcn_s_wait_asynccnt)
    __builtin_amdgcn_s_wait_asynccnt(2);
#else
    asm volatile("s_wait_asynccnt 2" ::: "memory");
#endif
}

// ---------------------------------------------------------------------------
// Kernel A: per-row squared norms. One wave per row, lane = float4.
// ---------------------------------------------------------------------------
__global__ __launch_bounds__(256)
void hc_sqnorm_kernel(const float* __restrict__ f, float* __restrict__ sq) {
    const int wave = threadIdx.x >> 5;
    const int lane = threadIdx.x & 31;
    const int row  = blockIdx.x * 8 + wave;
    const float4* fp = (const float4*)(f + row * DIMS);
    float4 a = fp[lane];
    float s = a.x * a.x + a.y * a.y + a.z * a.z + a.w * a.w;
    #pragma unroll
    for (int m = 16; m >= 1; m >>= 1) s += __shfl_xor(s, m, 32);
    if (lane == 0) sq[row] = s;
}

// ---------------------------------------------------------------------------
// Kernel B: fused WMMA distance tiles + streaming (min, sum-exp) reductions.
// Grid: 128 blocks = 32 row-blocks (128 rows) x 4 column splits, 256 threads.
// Wave w owns row-tile rb*8+w; all 8 waves share one async-staged B tile
// per iteration (double-buffered in LDS).
// ---------------------------------------------------------------------------
__global__ __launch_bounds__(256)
void hc_main_kernel(const float* __restrict__ f,
                    const float* __restrict__ sq,
                    float* __restrict__ partM, float* __restrict__ partS,
                    float* __restrict__ posT,  float* __restrict__ posW) {
    __shared__ float Blds[2][TILE * BSTRIDE];

    const int rb    = blockIdx.x >> 2;       // row block 0..31
    const int split = blockIdx.x & (NSPLIT - 1);
    const int tid  = threadIdx.x;
    const int wave = tid >> 5;
    const int lane = tid & 31;
    const int half = lane >> 4;              // 0: lanes 0-15, 1: lanes 16-31
    const int l16  = lane & 15;

    const int myRowTile = rb * NWAVES + wave;    // 0..255
    const int rowBase   = myRowTile * TILE;

    // B-tile staging coords: thread t copies two 16B chunks of column sc.
    const int sc = tid >> 4;                 // 0..15 column within tile
    const int sk = (tid & 15) * 8;           // float offset 0..120

    // Preload this wave's A tile (16x128) into registers: 32 x v2f.
    v2f areg[32];
    {
        const float* abase = f + (rowBase + l16) * DIMS + 2 * half;
        #pragma unroll
        for (int kk = 0; kk < 32; ++kk)
            areg[kk] = *(const v2f*)(abase + kk * 4);
    }

    // Row norms for this lane's 8 accumulator rows (VGPR i -> row i + 8*half).
    float rsq[8];
    #pragma unroll
    for (int i = 0; i < 8; ++i) rsq[i] = sq[rowBase + i + 8 * half];

    // Online (min, sum-exp) state per owned (row, column-slot) stream.
    float m[8], s[8];
    #pragma unroll
    for (int i = 0; i < 8; ++i) { m[i] = 1e30f; s[i] = 0.0f; }

    // Prologue: stage col-tile (split) into buffer 0.
    {
        const float* g = f + ((split * TILE) + sc) * DIMS + sk;
        float* l = &Blds[0][sc * BSTRIDE + sk];
        async_ld_b128(g, l);
        async_ld_b128(g + 4, l + 4);
    }

    for (int tt = 0; tt < TT_ITERS; ++tt) {
        const int ct      = split + NSPLIT * tt;
        const int colBase = ct * TILE;

        // Stage next tile into the other buffer (wraps; redundant last issue
        // keeps the ASYNCcnt invariant so wait<=2 always means "current done").
        {
            const int ctn = split + NSPLIT * ((tt + 1) & (TT_ITERS - 1));
            const float* g = f + (ctn * TILE + sc) * DIMS + sk;
            float* l = &Blds[(tt + 1) & 1][sc * BSTRIDE + sk];
            async_ld_b128(g, l);
            async_ld_b128(g + 4, l + 4);
        }
        wait_async_le2();        // current tile's 2 async ops complete
        __syncthreads();         // all waves' writes visible

        // Gram tile: 32 x V_WMMA_F32_16X16X4_F32, K = 128, B from LDS.
        const float* bb = &Blds[tt & 1][l16 * BSTRIDE + 2 * half];
        v8f acc = {};
        #pragma unroll
        for (int k0 = 0; k0 < DIMS; k0 += 4) {
            v2f b = *(const v2f*)(bb + k0);
            acc = __builtin_amdgcn_wmma_f32_16x16x4_f32(
                false, areg[k0 >> 2], false, b, (short)0, acc, false, false);
        }

        const float csq   = sq[colBase + l16];
        const int   gcolg = (colBase + l16) >> 2;     // column group id

        #pragma unroll
        for (int i = 0; i < 8; ++i) {
            const int grow = rowBase + i + 8 * half;
            const float d  = rsq[i] + csq - 2.0f * acc[i];
            const bool samegrp = ((grow >> 2) == gcolg);
            const float dn = samegrp ? 1e30f : d;     // mask -> exp underflows
            const float mn = fminf(m[i], dn);
            s[i] = s[i] * __expf(mn - m[i]) + __expf(mn - dn);
            m[i] = mn;
        }

        // Diagonal tile: extract the 3 positive distances per row.
        if (ct == myRowTile) {
            #pragma unroll
            for (int i = 0; i < 8; ++i) {
                const int rl = i + 8 * half;                     // local row
                const bool ispos = ((rl >> 2) == (l16 >> 2)) && (l16 != rl);
                const float d = rsq[i] + csq - 2.0f * acc[i];
                float pm = ispos ? d : -1e30f;
                #pragma unroll
                for (int msk = 8; msk >= 1; msk >>= 1)
                    pm = fmaxf(pm, __shfl_xor(pm, msk, 32));
                float pe = ispos ? __expf(d - pm) : 0.0f;
                #pragma unroll
                for (int msk = 8; msk >= 1; msk >>= 1)
                    pe += __shfl_xor(pe, msk, 32);
                if (l16 == 0) {
                    posT[rowBase + rl] = pm;
                    posW[rowBase + rl] = pe * (1.0f / 3.0f);
                }
            }
        }
        __syncthreads();         // readers done before buffer is overwritten
    }

    // Merge (m,s) across the 16 lanes of each half; write per-split partials.
    #pragma unroll
    for (int i = 0; i < 8; ++i) {
        #pragma unroll
        for (int msk = 1; msk <= 8; msk <<= 1) {
            const float m2 = __shfl_xor(m[i], msk, 32);
            const float s2 = __shfl_xor(s[i], msk, 32);
            const float mn = fminf(m[i], m2);
            s[i] = s[i] * __expf(mn - m[i]) + s2 * __expf(mn - m2);
            m[i] = mn;
        }
        if (l16 == 0) {
            const int row = rowBase + i + 8 * half;
            partM[split * B_ROWS + row] = m[i];
            partS[split * B_ROWS + row] = s[i];
        }
    }
}

// ---------------------------------------------------------------------------
// Kernel C: merge the 4 column-split partials per row, compute row loss.
// ---------------------------------------------------------------------------
__global__ __launch_bounds__(256)
void hc_rowmerge_kernel(const float* __restrict__ partM,
                        const float* __restrict__ partS,
                        const float* __restrict__ posT,
                        const float* __restrict__ posW,
                        float* __restrict__ rowLoss) {
    const int row = blockIdx.x * 256 + threadIdx.x;
    float mm = 1e30f, ss = 0.0f;
    #pragma unroll
    for (int p = 0; p < NSPLIT; ++p) {
        const float m2 = partM[p * B_ROWS + row];
        const float s2 = partS[p * B_ROWS + row];
        const float mn = fminf(mm, m2);
        ss = ss * __expf(mn - mm) + s2 * __expf(mn - m2);
        mm = mn;
    }
    const float between = ss * (1.0f / (float)(B_ROWS - EACH_CL));   // /4092
    const float gap  = posT[row] - mm;
    const float prod = 10000.0f * posW[row] * between;               // BETA
    float l;
    if (gap < 20.0f) l = log1pf(prod * __expf(gap));                 // MAX_GAP
    else             l = logf(prod) + gap;
    rowLoss[row] = l * (1.0f / 9.210440366976517f);                  // /log(1+BETA)
}

// ---------------------------------------------------------------------------
// Kernel D: mean over 4096 row losses.
// ---------------------------------------------------------------------------
__global__ __launch_bounds__(256)
void hc_reduce_kernel(const float* __restrict__ rowLoss, float* __restrict__ out) {
    __shared__ float red[256];
    float s = 0.0f;
    for (int i = threadIdx.x; i < B_ROWS; i += 256) s += rowLoss[i];
    red[threadIdx.x] = s;
    __syncthreads();
    #pragma unroll
    for (int st = 128; st > 0; st >>= 1) {
        if ((int)threadIdx.x < st) red[threadIdx.x] += red[threadIdx.x + st];
        __syncthreads();
    }
    if (threadIdx.x == 0) out[0] = red[0] * (1.0f / (float)B_ROWS);
}

extern "C" void kernel_launch(void* const* d_in, const int* in_sizes, int n_in,
                              void* d_out, int out_size, void* d_ws, size_t ws_size,
                              hipStream_t stream) {
    const float* feature = (const float*)d_in[0];   // (4096, 128) fp32
    // d_in[1] = label (unused by the loss)
    float* out = (float*)d_out;

    float* sq      = (float*)d_ws;                       // 4096
    float* partM   = sq      + B_ROWS;                   // 4 * 4096
    float* partS   = partM   + NSPLIT * B_ROWS;          // 4 * 4096
    float* posT    = partS   + NSPLIT * B_ROWS;          // 4096
    float* posW    = posT    + B_ROWS;                   // 4096
    float* rowLoss = posW    + B_ROWS;                   // 4096

    hc_sqnorm_kernel  <<<B_ROWS / 8,        256, 0, stream>>>(feature, sq);
    hc_main_kernel    <<<32 * NSPLIT,       256, 0, stream>>>(feature, sq,
                                                              partM, partS, posT, posW);
    hc_rowmerge_kernel<<<B_ROWS / 256,      256, 0, stream>>>(partM, partS,
                                                              posT, posW, rowLoss);
    hc_reduce_kernel  <<<1,                 256, 0, stream>>>(rowLoss, out);
}